// ItemGCN_84550726189738
// MI455X (gfx1250) — compile-verified
//
#include <hip/hip_runtime.h>
#include <hip/hip_bf16.h>

#define N_NODES 50000
#define DIM     256
#define HEADS   8
#define HDIM    32
#define NEDGE   500000
#define ETOT    (NEDGE + N_NODES)   // self-loops appended
#define NEGS    0.2f

typedef __attribute__((ext_vector_type(2))) float v2f;
typedef __attribute__((ext_vector_type(8))) float v8f;

// ---------------- fp32 WMMA GEMM: C[N,256] = A[N,256] @ W[256,256] (+ bias) ---
// 512 threads = 16 waves. Block bx owns rows [80*bx, 80*bx+80) (5 row tiles).
// Wave w owns columns [16*w, 16*w+16). Each B fragment is reused across the 5
// row tiles -> 5x less W traffic. A slab staged in LDS (padded stride 260 so
// the 16x4 fragment reads hit distinct banks). 80*260*4 = 83200 B LDS (<320KB).
#define LDS_STRIDE 260
#define RTILES 5
#define ROWS_PER_BLOCK (RTILES * 16)    // 80; 50000/80 = 625 blocks exactly

__global__ __launch_bounds__(512)
void k_gemm_wmma(const float* __restrict__ A, const float* __restrict__ W,
                 const float* __restrict__ bias, float* __restrict__ C,
                 int useBias)
{
    __shared__ float As[ROWS_PER_BLOCK * LDS_STRIDE];
    const int tid = threadIdx.x;
    const int bx  = blockIdx.x;

    // stage 80x256 A slab into LDS (coalesced)
    const float* Ablk = A + (size_t)bx * ROWS_PER_BLOCK * DIM;
    #pragma unroll
    for (int i = 0; i < 40; ++i) {
        int idx = tid + i * 512;             // 0..20479
        int row = idx >> 8;
        int col = idx & 255;
        As[row * LDS_STRIDE + col] = Ablk[idx];
    }
    __syncthreads();

    const int wave = tid >> 5;
    const int lane = tid & 31;
    const int half = lane >> 4;              // 0 or 1
    const int l15  = lane & 15;
    const int kk   = half * 2;               // K sub-offset for this half-wave
    const int colg = wave * 16 + l15;        // global output column

    v8f acc[RTILES] = {};

    #pragma unroll 2
    for (int k0 = 0; k0 < DIM; k0 += 4) {
        // B fragment (4x16): lane l15 = col N, holds K = k0+kk, k0+kk+1
        v2f b;
        b.x = W[(size_t)(k0 + kk)     * DIM + colg];
        b.y = W[(size_t)(k0 + kk + 1) * DIM + colg];
        #pragma unroll
        for (int t = 0; t < RTILES; ++t) {
            // A fragment (16x4): lane l15 = row M, holds K = k0+kk, k0+kk+1
            v2f a;
            a.x = As[(t * 16 + l15) * LDS_STRIDE + k0 + kk];
            a.y = As[(t * 16 + l15) * LDS_STRIDE + k0 + kk + 1];
            acc[t] = __builtin_amdgcn_wmma_f32_16x16x4_f32(
                         false, a, false, b, (short)0, acc[t], false, false);
        }
    }

    const float bv = useBias ? bias[colg] : 0.0f;
    // C/D layout: VGPR r -> row M = half*8 + r, col N = l15
    #pragma unroll
    for (int t = 0; t < RTILES; ++t) {
        const size_t rowbase = (size_t)bx * ROWS_PER_BLOCK + t * 16 + half * 8;
        #pragma unroll
        for (int r = 0; r < 8; ++r)
            C[(rowbase + r) * DIM + colg] = acc[t][r] + bv;
    }
}

// ---------------- attention coefficients: el/er [N,H] -------------------------
__global__ __launch_bounds__(256)
void k_attn(const float* __restrict__ h, const float* __restrict__ al,
            const float* __restrict__ ar, float* __restrict__ el,
            float* __restrict__ er)
{
    int idx = blockIdx.x * 256 + threadIdx.x;          // idx = n*8 + head
    if (idx >= N_NODES * HEADS) return;
    int head = idx & 7;
    int n    = idx >> 3;
    const float4* hp  = (const float4*)(h + (size_t)n * DIM + head * HDIM);
    const float4* alp = (const float4*)(al + head * HDIM);
    const float4* arp = (const float4*)(ar + head * HDIM);
    float sl = 0.f, sr = 0.f;
    #pragma unroll
    for (int i = 0; i < 8; ++i) {
        float4 v = hp[i], a = alp[i], b = arp[i];
        sl += v.x * a.x + v.y * a.y + v.z * a.z + v.w * a.w;
        sr += v.x * b.x + v.y * b.y + v.z * b.z + v.w * b.w;
    }
    el[idx] = sl;
    er[idx] = sr;
}

// ---------------- monotonic float <-> uint for order-independent atomic max ---
__device__ __forceinline__ unsigned fflip(float f) {
    unsigned u = __float_as_uint(f);
    return (u & 0x80000000u) ? ~u : (u | 0x80000000u);
}
__device__ __forceinline__ float funflip(unsigned u) {
    unsigned v = (u & 0x80000000u) ? (u & 0x7FFFFFFFu) : ~u;
    return __uint_as_float(v);
}

__global__ __launch_bounds__(256)
void k_init_mz(unsigned* __restrict__ mu, float* __restrict__ z)
{
    int idx = blockIdx.x * 256 + threadIdx.x;
    if (idx >= N_NODES * HEADS) return;
    mu[idx] = fflip(-__builtin_inff());
    z[idx]  = 0.0f;
}

__global__ __launch_bounds__(256)
void k_init_out(const float* __restrict__ b, float* __restrict__ out)
{
    int idx = blockIdx.x * 256 + threadIdx.x;
    if (idx >= N_NODES * DIM) return;
    out[idx] = b[idx & (DIM - 1)];
}

__device__ __forceinline__ void edge_sd(int eid, const int* __restrict__ src,
                                        const int* __restrict__ dst,
                                        int& s, int& d)
{
    if (eid < NEDGE) { s = src[eid]; d = dst[eid]; }
    else             { s = eid - NEDGE; d = s; }     // self-loop
}

__device__ __forceinline__ float edge_e(const float* __restrict__ el,
                                        const float* __restrict__ er,
                                        int s, int d, int head)
{
    float v = el[s * HEADS + head] + er[d * HEADS + head];
    return v > 0.f ? v : NEGS * v;                   // leaky relu
}

// ---------------- edge pass 1: segment max -----------------------------------
__global__ __launch_bounds__(256)
void k_edge_max(const int* __restrict__ src, const int* __restrict__ dst,
                const float* __restrict__ el, const float* __restrict__ er,
                unsigned* __restrict__ mu)
{
    int idx = blockIdx.x * 256 + threadIdx.x;        // idx = eid*8 + head
    if (idx >= ETOT * HEADS) return;
    int head = idx & 7, eid = idx >> 3, s, d;
    edge_sd(eid, src, dst, s, d);
    float e = edge_e(el, er, s, d, head);
    atomicMax(&mu[d * HEADS + head], fflip(e));
}

// ---------------- edge pass 2: exp + segment sum ------------------------------
__global__ __launch_bounds__(256)
void k_edge_exp(const int* __restrict__ src, const int* __restrict__ dst,
                const float* __restrict__ el, const float* __restrict__ er,
                const unsigned* __restrict__ mu, float* __restrict__ z,
                float* __restrict__ ea)
{
    int idx = blockIdx.x * 256 + threadIdx.x;
    if (idx >= ETOT * HEADS) return;
    int head = idx & 7, eid = idx >> 3, s, d;
    edge_sd(eid, src, dst, s, d);
    float e = edge_e(el, er, s, d, head);
    float a = expf(e - funflip(mu[d * HEADS + head]));
    ea[idx] = a;
    atomicAdd(&z[d * HEADS + head], a);
}

// ---------------- edge pass 3: normalized scatter-aggregate -------------------
__global__ __launch_bounds__(256)
void k_edge_agg(const int* __restrict__ src, const int* __restrict__ dst,
                const float* __restrict__ ea, const float* __restrict__ z,
                const float* __restrict__ h, float* __restrict__ out)
{
    int idx = blockIdx.x * 256 + threadIdx.x;
    if (idx >= ETOT * HEADS) return;
    int head = idx & 7, eid = idx >> 3, s, d;
    edge_sd(eid, src, dst, s, d);
    float alpha = ea[idx] / z[d * HEADS + head];
    const float4* hp = (const float4*)(h + (size_t)s * DIM + head * HDIM);
    float* op = out + (size_t)d * DIM + head * HDIM;
    #pragma unroll
    for (int i = 0; i < 8; ++i) {
        float4 v = hp[i];
        atomicAdd(op + 4 * i + 0, alpha * v.x);
        atomicAdd(op + 4 * i + 1, alpha * v.y);
        atomicAdd(op + 4 * i + 2, alpha * v.z);
        atomicAdd(op + 4 * i + 3, alpha * v.w);
    }
}

// ---------------- elementwise leaky relu -------------------------------------
__global__ __launch_bounds__(256)
void k_leaky(const float* __restrict__ in, float* __restrict__ out)
{
    int idx = blockIdx.x * 256 + threadIdx.x;
    if (idx >= N_NODES * DIM) return;
    float v = in[idx];
    out[idx] = v > 0.f ? v : NEGS * v;
}

// =============================================================================
extern "C" void kernel_launch(void* const* d_in, const int* in_sizes, int n_in,
                              void* d_out, int out_size, void* d_ws, size_t ws_size,
                              hipStream_t stream)
{
    const float* feats  = (const float*)d_in[0];
    const int*   src    = (const int*)  d_in[1];
    const int*   dst    = (const int*)  d_in[2];
    const float* proj_W = (const float*)d_in[3];
    const float* proj_b = (const float*)d_in[4];
    const float* Ws[2]  = {(const float*)d_in[5], (const float*)d_in[9]};
    const float* als[2] = {(const float*)d_in[6], (const float*)d_in[10]};
    const float* ars[2] = {(const float*)d_in[7], (const float*)d_in[11]};
    const float* bs[2]  = {(const float*)d_in[8], (const float*)d_in[12]};
    float* outp = (float*)d_out;

    // workspace carve-up
    const size_t ND = (size_t)N_NODES * DIM;     // 12.8M floats
    const size_t NH = (size_t)N_NODES * HEADS;   // 0.4M
    const size_t EH = (size_t)ETOT * HEADS;      // 4.4M
    float*    xbuf = (float*)d_ws;
    float*    hbuf = xbuf + ND;
    float*    obuf = hbuf + ND;
    float*    el   = obuf + ND;
    float*    er   = el + NH;
    unsigned* mu   = (unsigned*)(er + NH);
    float*    z    = (float*)(mu + NH);
    float*    ea   = z + NH;
    (void)ws_size; (void)in_sizes; (void)n_in; (void)out_size;

    const int GEMM_BLOCKS = N_NODES / ROWS_PER_BLOCK;   // 625 exact
    const int NH_BLOCKS   = (int)((NH + 255) / 256);
    const int ND_BLOCKS   = (int)((ND + 255) / 256);
    const int EH_BLOCKS   = (int)((EH + 255) / 256);

    // x = feats @ proj_W + proj_b
    k_gemm_wmma<<<GEMM_BLOCKS, 512, 0, stream>>>(feats, proj_W, proj_b, xbuf, 1);

    for (int L = 0; L < 2; ++L) {
        float* agg = (L == 0) ? obuf : outp;

        // h = x @ W
        k_gemm_wmma<<<GEMM_BLOCKS, 512, 0, stream>>>(xbuf, Ws[L], nullptr, hbuf, 0);
        // el/er
        k_attn<<<NH_BLOCKS, 256, 0, stream>>>(hbuf, als[L], ars[L], el, er);
        // init reductions + bias-preloaded accumulator
        k_init_mz <<<NH_BLOCKS, 256, 0, stream>>>(mu, z);
        k_init_out<<<ND_BLOCKS, 256, 0, stream>>>(bs[L], agg);
        // edge softmax + aggregation
        k_edge_max<<<EH_BLOCKS, 256, 0, stream>>>(src, dst, el, er, mu);
        k_edge_exp<<<EH_BLOCKS, 256, 0, stream>>>(src, dst, el, er, mu, z, ea);
        k_edge_agg<<<EH_BLOCKS, 256, 0, stream>>>(src, dst, ea, z, hbuf, agg);
        // module leaky relu; layer-1 result becomes next x, layer-2 in place
        if (L == 0) k_leaky<<<ND_BLOCKS, 256, 0, stream>>>(agg, xbuf);
        else        k_leaky<<<ND_BLOCKS, 256, 0, stream>>>(agg, agg);
    }
}